// GNNNet_15951508538236
// MI455X (gfx1250) — compile-verified
//
#include <hip/hip_runtime.h>

#define N_NODES 16384
#define N_EDGES 262144
#define DIM 256
#define NODES_PER_G 512

typedef __attribute__((ext_vector_type(2))) float v2f;
typedef __attribute__((ext_vector_type(8))) float v8f;

// ---------------------------------------------------------------- utilities
__global__ __launch_bounds__(256) void zero_f32(float* __restrict__ p, int n) {
    const int i = blockIdx.x * 256 + threadIdx.x;
    if (i < n) p[i] = 0.0f;
}

__global__ __launch_bounds__(256) void degree_count(const long long* __restrict__ dst,
                                                    float* __restrict__ deg) {
    const int e = blockIdx.x * 256 + threadIdx.x;
    if (e < N_EDGES) atomicAdd(&deg[(int)dst[e]], 1.0f);
}

__global__ __launch_bounds__(256) void finalize_invdeg(float* __restrict__ deg) {
    const int i = blockIdx.x * 256 + threadIdx.x;
    if (i < N_NODES) deg[i] = 1.0f / fmaxf(deg[i], 1.0f);
}

// ------------------------------------------------- edge gather + scatter-add
// 64 threads (2 waves) per edge; lane-coalesced float4 gather of x[src],
// 4 f32 atomics into agg[dst] (L2-resident, 16MB).
__global__ __launch_bounds__(256) void edge_scatter_add(const float* __restrict__ x,
                                                        const long long* __restrict__ src,
                                                        const long long* __restrict__ dst,
                                                        float* __restrict__ agg) {
    const unsigned gid = blockIdx.x * 256u + threadIdx.x;
    const unsigned e = gid >> 6;
    const unsigned f4 = gid & 63u;
    if (e >= N_EDGES) return;
    const int s = (int)src[e];
    const int d = (int)dst[e];
    const float4 v = ((const float4*)(x + (size_t)s * DIM))[f4];
    float* o = agg + (size_t)d * DIM + (f4 << 2);
    atomicAdd(o + 0, v.x);
    atomicAdd(o + 1, v.y);
    atomicAdd(o + 2, v.z);
    atomicAdd(o + 3, v.w);
}

// --------------------------------------------------- dual fp32 WMMA GEMM
// y[16-row strip, all 256 cols] = diag(inv_deg)*(agg @ Wl) + bl + x @ Wr
// 256 thr = 8 waves; wave owns two 16x16 col tiles (nc0, nc0+128).
// Strips staged via async global->LDS copies (ASYNCcnt path).
__global__ __launch_bounds__(256) void sage_gemm_wmma(const float* __restrict__ x,
                                                      const float* __restrict__ agg,
                                                      const float* __restrict__ invdeg,
                                                      const float* __restrict__ Wl,
                                                      const float* __restrict__ Wr,
                                                      const float* __restrict__ bl,
                                                      float* __restrict__ y) {
    __shared__ float sA[16][DIM + 4];   // +4 pad: conflict-free A-frag reads
    __shared__ float sX[16][DIM + 4];
    const int tid  = threadIdx.x;
    const int lane = tid & 31;
    const int wave = tid >> 5;
    const int row0 = blockIdx.x << 4;
    const int nc0  = wave << 4;          // col tile 0
    const int nc1  = nc0 + 128;          // col tile 1
    const int half = lane >> 4;          // K sub-pair select (16x4 f32 A layout)
    const int l16  = lane & 15;

    // async-stage 16x256 strips of x and raw agg straight into LDS
    for (int i = tid; i < 16 * (DIM / 4); i += 256) {
        const int r  = i >> 6;
        const int c4 = i & 63;
        const float* gx = x   + (size_t)(row0 + r) * DIM + (c4 << 2);
        const float* ga = agg + (size_t)(row0 + r) * DIM + (c4 << 2);
        const unsigned lx = (unsigned)(uintptr_t)&sX[r][c4 << 2];
        const unsigned la = (unsigned)(uintptr_t)&sA[r][c4 << 2];
        asm volatile("global_load_async_to_lds_b128 %0, %1, off"
                     :: "v"(lx), "v"(gx) : "memory");
        asm volatile("global_load_async_to_lds_b128 %0, %1, off"
                     :: "v"(la), "v"(ga) : "memory");
    }
    asm volatile("s_wait_asynccnt 0x0" ::: "memory");
    __syncthreads();

    __builtin_prefetch(Wl + nc0 + l16, 0, 0);   // global_prefetch_b8: warm weights
    __builtin_prefetch(Wr + nc0 + l16, 0, 0);

    const float* sArow = &sA[l16][0];
    const float* sXrow = &sX[l16][0];
    v8f accA0 = {}, accX0 = {}, accA1 = {}, accX1 = {};   // 4 independent chains
    for (int k = 0; k < DIM; k += 4) {
        const int kk = k + 2 * half;
        v2f a, xf, b0l, b0r, b1l, b1r;
        a.x  = sArow[kk]; a.y  = sArow[kk + 1];
        xf.x = sXrow[kk]; xf.y = sXrow[kk + 1];
        const float* wl = Wl + (size_t)kk * DIM + nc0 + l16;
        const float* wr = Wr + (size_t)kk * DIM + nc0 + l16;
        b0l.x = wl[0];   b0l.y = wl[DIM];
        b0r.x = wr[0];   b0r.y = wr[DIM];
        b1l.x = wl[128]; b1l.y = wl[DIM + 128];
        b1r.x = wr[128]; b1r.y = wr[DIM + 128];
        accA0 = __builtin_amdgcn_wmma_f32_16x16x4_f32(false, a,  false, b0l, (short)0, accA0, false, false);
        accX0 = __builtin_amdgcn_wmma_f32_16x16x4_f32(false, xf, false, b0r, (short)0, accX0, false, false);
        accA1 = __builtin_amdgcn_wmma_f32_16x16x4_f32(false, a,  false, b1l, (short)0, accA1, false, false);
        accX1 = __builtin_amdgcn_wmma_f32_16x16x4_f32(false, xf, false, b1r, (short)0, accX1, false, false);
    }
    // C/D layout: VGPR r holds M=r (lanes 0-15) / M=r+8 (lanes 16-31), N=l16.
    // Fold mean-aggregation scale here: diag(s)*A @ W == diag(s)*(A @ W).
    const float bias0 = bl[nc0 + l16];
    const float bias1 = bl[nc1 + l16];
    const int grow = row0 + 8 * half;
    float* yp = y + (size_t)grow * DIM + nc0 + l16;
#pragma unroll
    for (int r = 0; r < 8; ++r) {
        const float s = invdeg[grow + r];
        yp[(size_t)r * DIM]       = accA0[r] * s + accX0[r] + bias0;
        yp[(size_t)r * DIM + 128] = accA1[r] * s + accX1[r] + bias1;
    }
}

// --------------------------------------------------- tiny MLP head (32x256)
__global__ __launch_bounds__(256) void mlp_head_wmma(const float* __restrict__ xin,
                                                     const float* __restrict__ Wh,
                                                     const float* __restrict__ bh,
                                                     const float* __restrict__ Wo,
                                                     const float* __restrict__ bo,
                                                     float* __restrict__ out) {
    __shared__ float h[32][DIM + 4];
    const int tid  = threadIdx.x;
    const int lane = tid & 31;
    const int wave = tid >> 5;
    const int half = lane >> 4;
    const int l16  = lane & 15;

    // gather root node of each graph: node g*512
    for (int i = tid; i < 32 * (DIM / 4); i += 256) {
        const int g  = i >> 6;
        const int c4 = i & 63;
        ((float4*)&h[g][0])[c4] = ((const float4*)(xin + (size_t)g * NODES_PER_G * DIM))[c4];
    }
    __syncthreads();

    for (int layer = 0; layer < 3; ++layer) {
        const float* W  = (layer < 2) ? (Wh + (size_t)layer * DIM * DIM) : Wo;
        const float* bv = (layer < 2) ? (bh + (size_t)layer * DIM) : bo;
        v8f acc[4] = {};
#pragma unroll
        for (int i = 0; i < 4; ++i) {
            const int t    = wave + (i << 3);       // 32 tiles: 2 Mtiles x 16 Ntiles
            const int mrow = (t >> 4) << 4;
            const int nc   = (t & 15) << 4;
            const float* hrow = &h[mrow + l16][0];
            for (int k = 0; k < DIM; k += 4) {
                const int kk = k + 2 * half;
                v2f a, b;
                a.x = hrow[kk]; a.y = hrow[kk + 1];
                const float* wp = W + (size_t)kk * DIM + nc + l16;
                b.x = wp[0]; b.y = wp[DIM];
                acc[i] = __builtin_amdgcn_wmma_f32_16x16x4_f32(false, a, false, b, (short)0, acc[i], false, false);
            }
        }
        __syncthreads();  // all h reads complete before overwrite
#pragma unroll
        for (int i = 0; i < 4; ++i) {
            const int t    = wave + (i << 3);
            const int mrow = (t >> 4) << 4;
            const int nc   = (t & 15) << 4;
            const float bb = bv[nc + l16];
#pragma unroll
            for (int r = 0; r < 8; ++r) {
                const float vv = acc[i][r] + bb;
                if (layer < 2)
                    h[mrow + r + 8 * half][nc + l16] = fmaxf(vv, 0.0f);
                else
                    out[(size_t)(mrow + r + 8 * half) * DIM + nc + l16] = vv;
            }
        }
        __syncthreads();
    }
}

// ----------------------------------------------------------------- launcher
extern "C" void kernel_launch(void* const* d_in, const int* in_sizes, int n_in,
                              void* d_out, int out_size, void* d_ws, size_t ws_size,
                              hipStream_t stream) {
    const float*     x  = (const float*)d_in[0];
    const long long* ei = (const long long*)d_in[1];   // int64 [2, E]
    const float*     Wl = (const float*)d_in[2];
    const float*     bl = (const float*)d_in[3];
    const float*     Wr = (const float*)d_in[4];
    const float*     Wh = (const float*)d_in[5];
    const float*     bh = (const float*)d_in[6];
    const float*     Wo = (const float*)d_in[7];
    const float*     bo = (const float*)d_in[8];
    float* out = (float*)d_out;

    char* ws = (char*)d_ws;
    const size_t FEAT_BYTES = (size_t)N_NODES * DIM * sizeof(float);
    float* agg    = (float*)(ws);
    float* xb0    = (float*)(ws + FEAT_BYTES);
    float* xb1    = (float*)(ws + 2 * FEAT_BYTES);
    float* invdeg = (float*)(ws + 3 * FEAT_BYTES);

    const long long* src = ei;
    const long long* dst = ei + N_EDGES;

    zero_f32<<<N_NODES / 256, 256, 0, stream>>>(invdeg, N_NODES);
    degree_count<<<N_EDGES / 256, 256, 0, stream>>>(dst, invdeg);
    finalize_invdeg<<<N_NODES / 256, 256, 0, stream>>>(invdeg);

    const float* cur = x;
    float* bufs[2] = {xb0, xb1};
    for (int l = 0; l < 3; ++l) {
        zero_f32<<<(N_NODES * DIM) / 256, 256, 0, stream>>>(agg, N_NODES * DIM);
        edge_scatter_add<<<(N_EDGES * 64) / 256, 256, 0, stream>>>(cur, src, dst, agg);
        sage_gemm_wmma<<<dim3(N_NODES / 16), 256, 0, stream>>>(cur, agg, invdeg,
                                                               Wl + (size_t)l * DIM * DIM,
                                                               Wr + (size_t)l * DIM * DIM,
                                                               bl + (size_t)l * DIM,
                                                               bufs[l & 1]);
        cur = bufs[l & 1];
    }
    mlp_head_wmma<<<1, 256, 0, stream>>>(cur, Wh, bh, Wo, bo, out);
}